// VectorQuantizerEMA_78314433675996
// MI455X (gfx1250) — compile-verified
//
#include <hip/hip_runtime.h>
#include <hip/hip_bf16.h>

typedef float v2f __attribute__((ext_vector_type(2)));
typedef float v8f __attribute__((ext_vector_type(8)));

#define N_PTS   200000
#define K_EMB   4096
#define D_DIM   256
#define DECAY   0.99f
#define BN_EPS  1e-05f
#define BN_MOM  0.1f
#define LAP_EPS 1e-05f

// ---- workspace layout (in 4-byte words) ----
#define WS_SUM     0        // 256 col sums
#define WS_SUMSQ   256      // 256 col sums of squares
#define WS_CNT     512      // 4096 int counts
#define WS_CURSOR  4608     // 4096 int cursors
#define WS_ZEROLEN 8704
#define WS_MEAN    8704     // 256
#define WS_RSTD    8960     // 256
#define WS_RMEAN   9216     // 256
#define WS_RSTDRUN 9472     // 256
#define WS_CS      9728     // 4096
#define WS_OFFS    13824    // 4097 int offsets

// ---------------- K0: zero the accumulator region ----------------
__global__ void k0_zero(float* __restrict__ ws) {
    int i = blockIdx.x * blockDim.x + threadIdx.x;
    if (i < WS_ZEROLEN) ws[i] = 0.0f;
}

// ---------------- K1: column sum / sumsq via WMMA ones-matmul, b128 loads ------
// Wave owns a 64-column group. Each lane loads float4 (4 consecutive columns)
// from two rows -> 2x global_load_b128 per iteration (1 KB/wave). Four WMMA
// accumulator chains (one per float4 component) compute column sums under the
// fixed permutation  logical n -> physical 4n+j ; four more chains accumulate
// sums of squares. D = ones(16x4) x B + D  => every lane's c[0] = column total.
// N_PTS % 4 == 0 and whole-chunk assignment => no per-load guards.
__global__ void k1_colstats_wmma(const float* __restrict__ x,
                                 float* __restrict__ ws) {
    const int lane       = threadIdx.x & 31;
    const int wavesPerBk = blockDim.x >> 5;
    const int wave       = blockIdx.x * wavesPerBk + (threadIdx.x >> 5);
    const int totalWaves = gridDim.x * wavesPerBk;      // multiple of 4
    const int ct         = wave & 3;                    // 64-column group 0..3
    const int wslot      = wave >> 2;
    const int wavesGrp   = totalWaves >> 2;
    const int cbase      = ct * 64 + 4 * (lane & 15);   // lane's 4 columns

    // number of 4-row chunks this wave slot owns (chunk = wslot + t*wavesGrp)
    const int nChunks = (N_PTS / 4 - wslot + wavesGrp - 1) / wavesGrp;

    // lanes 0-15: rows rbase+0/+1 ; lanes 16-31: rows rbase+2/+3
    const int ra = wslot * 4 + ((lane >> 4) << 1);
    const float4* __restrict__ p =
        (const float4*)(x + (size_t)ra * D_DIM + cbase);
    const size_t pstep = (size_t)wavesGrp * 4 * (D_DIM / 4); // float4 units

    v8f cs0 = {}, cs1 = {}, cs2 = {}, cs3 = {};   // column sums
    v8f cq0 = {}, cq1 = {}, cq2 = {}, cq3 = {};   // column sums of squares
    const v2f a = {1.0f, 1.0f};                   // all-ones A (16x4)

    #pragma unroll 2
    for (int t = 0; t < nChunks; ++t) {
        const float4 ua = p[0];            // row ra
        const float4 ub = p[D_DIM / 4];    // row ra+1
        p += pstep;
        v2f b0 = {ua.x, ub.x}, b1 = {ua.y, ub.y};
        v2f b2 = {ua.z, ub.z}, b3 = {ua.w, ub.w};
        v2f q0 = {ua.x * ua.x, ub.x * ub.x};
        v2f q1 = {ua.y * ua.y, ub.y * ub.y};
        v2f q2 = {ua.z * ua.z, ub.z * ub.z};
        v2f q3 = {ua.w * ua.w, ub.w * ub.w};
        cs0 = __builtin_amdgcn_wmma_f32_16x16x4_f32(false, a, false, b0, (short)0, cs0, false, false);
        cs1 = __builtin_amdgcn_wmma_f32_16x16x4_f32(false, a, false, b1, (short)0, cs1, false, false);
        cs2 = __builtin_amdgcn_wmma_f32_16x16x4_f32(false, a, false, b2, (short)0, cs2, false, false);
        cs3 = __builtin_amdgcn_wmma_f32_16x16x4_f32(false, a, false, b3, (short)0, cs3, false, false);
        cq0 = __builtin_amdgcn_wmma_f32_16x16x4_f32(false, a, false, q0, (short)0, cq0, false, false);
        cq1 = __builtin_amdgcn_wmma_f32_16x16x4_f32(false, a, false, q1, (short)0, cq1, false, false);
        cq2 = __builtin_amdgcn_wmma_f32_16x16x4_f32(false, a, false, q2, (short)0, cq2, false, false);
        cq3 = __builtin_amdgcn_wmma_f32_16x16x4_f32(false, a, false, q3, (short)0, cq3, false, false);
    }

    // every lane's c[0] holds the total for its column; lanes 0-15 commit once
    if (lane < 16) {
        float* wsum = ws + WS_SUM;
        float* wsq  = ws + WS_SUMSQ;
        atomicAdd(&wsum[cbase + 0], cs0[0]);
        atomicAdd(&wsum[cbase + 1], cs1[0]);
        atomicAdd(&wsum[cbase + 2], cs2[0]);
        atomicAdd(&wsum[cbase + 3], cs3[0]);
        atomicAdd(&wsq[cbase + 0], cq0[0]);
        atomicAdd(&wsq[cbase + 1], cq1[0]);
        atomicAdd(&wsq[cbase + 2], cq2[0]);
        atomicAdd(&wsq[cbase + 3], cq3[0]);
    }
}

// ---------------- K2: finalize per-column stats (1 block, 256 threads) ----------------
__global__ void k2_finalize_stats(const float* __restrict__ running_mean,
                                  const float* __restrict__ running_var,
                                  float* __restrict__ ws) {
    const int d = threadIdx.x;
    const float nf   = (float)N_PTS;
    const float s    = ws[WS_SUM + d];
    const float q    = ws[WS_SUMSQ + d];
    const float mean = s / nf;
    const float var  = q / nf - mean * mean;
    const float rstd = rsqrtf(var + BN_EPS);
    const float unb  = var * (nf / (nf - 1.0f));
    const float nrm  = (1.0f - BN_MOM) * running_mean[d] + BN_MOM * mean;
    const float nrv  = (1.0f - BN_MOM) * running_var[d]  + BN_MOM * unb;
    ws[WS_MEAN    + d] = mean;
    ws[WS_RSTD    + d] = rstd;
    ws[WS_RMEAN   + d] = nrm;
    ws[WS_RSTDRUN + d] = sqrtf(nrv + BN_EPS);
}

// ---------------- K3a: histogram of cluster ids ----------------
__global__ void k3a_count(const int* __restrict__ nodes, float* __restrict__ ws) {
    int* cnt = (int*)(ws + WS_CNT);
    const int i = blockIdx.x * blockDim.x + threadIdx.x;
    if (i < N_PTS) atomicAdd(&cnt[nodes[i]], 1);
}

// ---------------- K3b: scan over K=4096 + smoothed cluster sizes (1 block, 1024 thr) ---
__global__ void k3b_scan_cs(const float* __restrict__ ema_cs,
                            float* __restrict__ ws,
                            float* __restrict__ out_cs) {
    __shared__ int   sscan[1024];
    __shared__ float fscan[1024];
    const int* cnt  = (const int*)(ws + WS_CNT);
    int*       offs = (int*)(ws + WS_OFFS);
    float*     wcs  = ws + WS_CS;
    const int t = threadIdx.x;

    const int c0 = cnt[4 * t + 0], c1 = cnt[4 * t + 1];
    const int c2 = cnt[4 * t + 2], c3 = cnt[4 * t + 3];
    const int s  = c0 + c1 + c2 + c3;

    float r0 = ema_cs[4 * t + 0] * DECAY + (1.0f - DECAY) * (float)c0;
    float r1 = ema_cs[4 * t + 1] * DECAY + (1.0f - DECAY) * (float)c1;
    float r2 = ema_cs[4 * t + 2] * DECAY + (1.0f - DECAY) * (float)c2;
    float r3 = ema_cs[4 * t + 3] * DECAY + (1.0f - DECAY) * (float)c3;

    sscan[t] = s;
    fscan[t] = r0 + r1 + r2 + r3;
    __syncthreads();
    for (int off = 1; off < 1024; off <<= 1) {
        int v = 0; float f = 0.0f;
        if (t >= off) { v = sscan[t - off]; f = fscan[t - off]; }
        __syncthreads();
        sscan[t] += v; fscan[t] += f;
        __syncthreads();
    }
    const int ex = sscan[t] - s;                 // exclusive prefix of counts
    offs[4 * t + 0] = ex;
    offs[4 * t + 1] = ex + c0;
    offs[4 * t + 2] = ex + c0 + c1;
    offs[4 * t + 3] = ex + c0 + c1 + c2;
    if (t == 1023) offs[K_EMB] = sscan[1023];    // == N

    const float n_total = fscan[1023];
    const float scale   = n_total / (n_total + (float)K_EMB * LAP_EPS);
    const float v0 = (r0 + LAP_EPS) * scale;
    const float v1 = (r1 + LAP_EPS) * scale;
    const float v2 = (r2 + LAP_EPS) * scale;
    const float v3 = (r3 + LAP_EPS) * scale;
    wcs[4 * t + 0] = v0; out_cs[4 * t + 0] = v0;
    wcs[4 * t + 1] = v1; out_cs[4 * t + 1] = v1;
    wcs[4 * t + 2] = v2; out_cs[4 * t + 2] = v2;
    wcs[4 * t + 3] = v3; out_cs[4 * t + 3] = v3;
}

// ---------------- K3c: build row permutation grouped by cluster ----------------
__global__ void k3c_place(const int* __restrict__ nodes, float* __restrict__ ws,
                          int* __restrict__ perm) {
    const int* offs   = (const int*)(ws + WS_OFFS);
    int*       cursor = (int*)(ws + WS_CURSOR);
    const int i = blockIdx.x * blockDim.x + threadIdx.x;
    if (i < N_PTS) {
        const int c   = nodes[i];
        const int pos = offs[c] + atomicAdd(&cursor[c], 1);
        perm[pos] = i;
    }
}

// ---------------- K4: per-cluster gather-reduce + fused EMA / embedding output ----
// dw[c][d] = rstd[d] * (sum_rows x - cnt*mean[d]); new_ema_w = 0.99*ema_w + 0.01*dw;
// emb_out = (new_ema_w / cs[c]) * run_std[d] + run_mean[d]
__global__ void k4_gather_finalize(const float* __restrict__ x,
                                   const float* __restrict__ ema_w,
                                   const int*   __restrict__ perm,
                                   const float* __restrict__ ws,
                                   float* __restrict__ out_emb,
                                   float* __restrict__ out_ema) {
    const int c = blockIdx.x;
    const int d = threadIdx.x;
    const int* offs = (const int*)(ws + WS_OFFS);
    const float m   = ws[WS_MEAN + d];
    const float rs  = ws[WS_RSTD + d];
    const int start = offs[c];
    const int end   = offs[c + 1];

    float acc = 0.0f;
    int j = start;
    for (; j + 4 <= end; j += 4) {
        const int q0 = perm[j + 0], q1 = perm[j + 1];
        const int q2 = perm[j + 2], q3 = perm[j + 3];
        acc += x[(size_t)q0 * D_DIM + d];
        acc += x[(size_t)q1 * D_DIM + d];
        acc += x[(size_t)q2 * D_DIM + d];
        acc += x[(size_t)q3 * D_DIM + d];
    }
    for (; j < end; ++j) acc += x[(size_t)perm[j] * D_DIM + d];

    const float cntf = (float)(end - start);
    const float dw   = (acc - cntf * m) * rs;
    const float nw   = ema_w[(size_t)c * D_DIM + d] * DECAY + (1.0f - DECAY) * dw;
    out_ema[(size_t)c * D_DIM + d] = nw;
    const float emb = nw / ws[WS_CS + c];
    out_emb[(size_t)c * D_DIM + d] = emb * ws[WS_RSTDRUN + d] + ws[WS_RMEAN + d];
}

// ---------------- K5: overwrite perm region with float encoding indices ----------
__global__ void k5_indices(const int* __restrict__ nodes, float* __restrict__ out_idx) {
    const int i = blockIdx.x * blockDim.x + threadIdx.x;
    if (i < N_PTS) out_idx[i] = (float)nodes[i];
}

extern "C" void kernel_launch(void* const* d_in, const int* in_sizes, int n_in,
                              void* d_out, int out_size, void* d_ws, size_t ws_size,
                              hipStream_t stream) {
    const float* x        = (const float*)d_in[0];
    const int*   nodes    = (const int*)  d_in[1];
    const float* ema_cs   = (const float*)d_in[2];
    const float* ema_w    = (const float*)d_in[3];
    const float* run_mean = (const float*)d_in[4];
    const float* run_var  = (const float*)d_in[5];

    float* ws  = (float*)d_ws;
    float* out = (float*)d_out;
    float* out_emb = out;                                   // K*D
    float* out_ema = out + (size_t)K_EMB * D_DIM;           // K*D
    float* out_cs  = out + 2 * (size_t)K_EMB * D_DIM;       // K
    float* out_idx = out_cs + K_EMB;                        // N  (also perm scratch)
    int*   perm    = (int*)out_idx;

    const int nBlkN = (N_PTS + 255) / 256;

    k0_zero           <<<(WS_ZEROLEN + 255) / 256, 256, 0, stream>>>(ws);
    k1_colstats_wmma  <<<512, 256, 0, stream>>>(x, ws);
    k2_finalize_stats <<<1, 256, 0, stream>>>(run_mean, run_var, ws);
    k3a_count         <<<nBlkN, 256, 0, stream>>>(nodes, ws);
    k3b_scan_cs       <<<1, 1024, 0, stream>>>(ema_cs, ws, out_cs);
    k3c_place         <<<nBlkN, 256, 0, stream>>>(nodes, ws, perm);
    k4_gather_finalize<<<K_EMB, D_DIM, 0, stream>>>(x, ema_w, perm, ws, out_emb, out_ema);
    k5_indices        <<<nBlkN, 256, 0, stream>>>(nodes, out_idx);
}